// VectorQuantizerEMA_42013370089749
// MI455X (gfx1250) — compile-verified
//
#include <hip/hip_runtime.h>
#include <hip/hip_bf16.h>
#include <math.h>

// ---------------------------------------------------------------------------
// VQ-VAE quantizer forward for MI455X (gfx1250, wave32).
//   z:   [16,4096,256] f32  -> N=65536 rows, D=256
//   E:   [256, 8192]   f32  -> K=8192 codes (column-major codes)
// Fused f32-WMMA GEMM + argmin; [N,K] distances never materialized.
// Double-buffered LDS B-tiles: next tile's global loads issue before the
// 64-WMMA burst (latency hidden under the matrix pipe), one barrier/tile.
// ---------------------------------------------------------------------------

typedef __attribute__((ext_vector_type(2))) float v2f;
typedef __attribute__((ext_vector_type(8))) float v8f;

#define D_DIM 256
#define K_CODES 8192
#define N_ROWS 65536
#define ROWS_PER_BLOCK 128                    // 8 waves x 16 rows
#define NUM_BLOCKS (N_ROWS / ROWS_PER_BLOCK)  // 512
#define EB_STRIDE 260   // 256 + 4 pad floats: conflict-free ds_load_2addr_b64

// d_out layout (f32, concatenated tuple):
#define ZST_OFF  0
#define LOSS_OFF (16 * 4096 * 256)            // 16777216
#define IDX_OFF  (LOSS_OFF + 1)
#define PPL_OFF  (IDX_OFF + N_ROWS)
#define USE_OFF  (PPL_OFF + 1)

// ---------------------------------------------------------------------------
__global__ __launch_bounds__(256) void vq_zero_counts(int* __restrict__ counts) {
    counts[blockIdx.x * 256 + threadIdx.x] = 0;
}

// e_sq[k] = sum_d E[d,k]^2   (lanes take consecutive k -> coalesced)
__global__ __launch_bounds__(256) void vq_esq(const float* __restrict__ emb,
                                              float* __restrict__ esq) {
    int k = blockIdx.x * 256 + threadIdx.x;
    float s = 0.f;
#pragma unroll 8
    for (int d = 0; d < D_DIM; ++d) {
        float v = emb[(size_t)d * K_CODES + k];
        s = __builtin_fmaf(v, v, s);
    }
    esq[k] = s;
}

// ---------------------------------------------------------------------------
// helpers for the main kernel
// ---------------------------------------------------------------------------
__device__ __forceinline__ void vq_load_tile(float stg[16], const float* __restrict__ emb,
                                             int k0, int jcode, int dblk) {
    const float* src = emb + (size_t)(dblk * 16) * K_CODES + (k0 + jcode);
#pragma unroll
    for (int i = 0; i < 16; ++i) stg[i] = src[(size_t)i * K_CODES];
}

__device__ __forceinline__ void vq_store_tile(float* __restrict__ buf, const float stg[16],
                                              int jcode, int dblk) {
    float* dst = buf + jcode * EB_STRIDE + dblk * 16;
#pragma unroll
    for (int i = 0; i < 16; ++i) dst[i] = stg[i];   // vectorizes to ds_store_b128
}

__device__ __forceinline__ void vq_wmma_tile(const float* __restrict__ buf,
                                             const v2f* __restrict__ a, v8f& c,
                                             int l15, int half) {
#pragma unroll
    for (int dc = 0; dc < 64; ++dc) {
        v2f b = *(const v2f*)&buf[l15 * EB_STRIDE + dc * 4 + half * 2];
        c = __builtin_amdgcn_wmma_f32_16x16x4_f32(
                false, a[dc], false, b, (short)0, c, false, false);
    }
}

__device__ __forceinline__ void vq_min_update(const v8f& c, const float* __restrict__ esq,
                                              int k0, int l15,
                                              float* __restrict__ best,
                                              int* __restrict__ bidx) {
    const float es   = esq[k0 + l15];
    const int   kidx = k0 + l15;
#pragma unroll
    for (int r = 0; r < 8; ++r) {
        float score = __builtin_fmaf(-2.f, c[r], es);
        if (score < best[r]) { best[r] = score; bidx[r] = kidx; }   // keep earliest k
    }
}

// ---------------------------------------------------------------------------
// Main fused GEMM + argmin kernel.
// Block = 256 threads = 8 waves; wave w owns rows [row0, row0+16).
// A (16x256 z tile) resident in 64 v2f registers in WMMA 16x16x4 A-layout:
//   lane l (l&15 = M), half = l>>4: chunk dc holds z[M, 4dc + 2*half + {0,1}].
// B tiles (E[0:256, k0:k0+16] transposed, eB[code][depth], stride 260) are
// double-buffered in LDS so staging overlaps the WMMA bursts.
// ---------------------------------------------------------------------------
__global__ __launch_bounds__(256) void vq_argmin_wmma(
    const float* __restrict__ z, const float* __restrict__ emb,
    const float* __restrict__ esq,
    int* __restrict__ codes, int* __restrict__ counts,
    float* __restrict__ partial_loss, float* __restrict__ out_indices) {

    __shared__ float eB0[16 * EB_STRIDE];   // double buffer, ~16.6 KB each
    __shared__ float eB1[16 * EB_STRIDE];
    __shared__ float zsq_s[8][16];
    __shared__ float wave_loss[16];

    const int tid  = threadIdx.x;
    const int lane = tid & 31;
    const int wave = tid >> 5;
    const int half = lane >> 4;     // 0 / 1
    const int l15  = lane & 15;
    const int row0 = blockIdx.x * ROWS_PER_BLOCK + wave * 16;

    // ---- preload A tile into registers (reused for all 512 K-tiles) ----
    v2f a[64];
    {
        const float* zrow = z + (size_t)(row0 + l15) * D_DIM + half * 2;
#pragma unroll
        for (int dc = 0; dc < 64; ++dc)
            a[dc] = *(const v2f*)(zrow + dc * 4);
    }

    // ---- ||z_row||^2 from resident A (half-lane partial + swap) ----
    float zp = 0.f;
#pragma unroll
    for (int dc = 0; dc < 64; ++dc)
        zp += a[dc].x * a[dc].x + a[dc].y * a[dc].y;
    zp += __shfl_xor(zp, 16, 32);          // combine column halves
    if (half == 0) zsq_s[wave][l15] = zp;  // row row0+l15

    float best[8];
    int   bidx[8];
#pragma unroll
    for (int r = 0; r < 8; ++r) { best[r] = 3.4e38f; bidx[r] = 0; }

    const int jcode = tid & 15;   // staging: code within tile
    const int dblk  = tid >> 4;   // staging: 16-depth block

    // ---- prologue: stage tile k0=0 into eB0 ----
    float stg[16];
    vq_load_tile(stg, emb, 0, jcode, dblk);
    vq_store_tile(eB0, stg, jcode, dblk);
    __syncthreads();

    // ---- main loop, 2 tiles per iteration (fixed buffers -> immediate ds offsets) ----
    for (int k0 = 0; k0 < K_CODES; k0 += 32) {
        // phase 0: compute from eB0, stage k0+16 into eB1 (always exists: K%32==0)
        vq_load_tile(stg, emb, k0 + 16, jcode, dblk);   // latency hides under WMMAs
        {
            v8f c = {};
            vq_wmma_tile(eB0, a, c, l15, half);
            vq_min_update(c, esq, k0, l15, best, bidx);
        }
        vq_store_tile(eB1, stg, jcode, dblk);
        __syncthreads();

        // phase 1: compute from eB1, stage k0+32 into eB0 (unless last)
        const bool more = (k0 + 32) < K_CODES;          // uniform branch, EXEC stays full
        if (more) vq_load_tile(stg, emb, k0 + 32, jcode, dblk);
        {
            v8f c = {};
            vq_wmma_tile(eB1, a, c, l15, half);
            vq_min_update(c, esq, k0 + 16, l15, best, bidx);
        }
        if (more) vq_store_tile(eB0, stg, jcode, dblk);
        __syncthreads();
    }

    // ---- cross-lane argmin within each 16-lane half (lane l&15 = k residue) ----
    float lsum = 0.f;
#pragma unroll
    for (int r = 0; r < 8; ++r) {
        float bd = best[r];
        int   bi = bidx[r];
#pragma unroll
        for (int m = 1; m < 16; m <<= 1) {
            float od = __shfl_xor(bd, m, 16);
            int   oi = __shfl_xor(bi, m, 16);
            if (od < bd || (od == bd && oi < bi)) { bd = od; bi = oi; }  // argmin-first ties
        }
        if (l15 == 0) {   // lanes 0 / 16 own rows row0+r / row0+8+r
            const int mrow = row0 + r + 8 * half;
            codes[mrow]       = bi;
            out_indices[mrow] = (float)bi;
            atomicAdd(&counts[bi], 1);
            lsum += zsq_s[wave][r + 8 * half] + bd;   // true squared distance
        }
    }

    // ---- deterministic per-block loss partial ----
    if (l15 == 0) wave_loss[wave * 2 + half] = lsum;
    __syncthreads();
    if (tid == 0) {
        float t = 0.f;
#pragma unroll
        for (int i = 0; i < 16; ++i) t += wave_loss[i];
        partial_loss[blockIdx.x] = t;
    }
}

// ---------------------------------------------------------------------------
// z_st = z_q + (z - z_q), computed with the reference's own op order.
// One float4 (4 depth elems) per thread; all lanes of a row share the code,
// so the strided embedding gather broadcasts out of L2 (E is L2-resident).
// ---------------------------------------------------------------------------
__global__ __launch_bounds__(256) void vq_writeback(
    const float* __restrict__ z, const float* __restrict__ emb,
    const int* __restrict__ codes, float* __restrict__ zst) {
    const int i4   = blockIdx.x * 256 + threadIdx.x;   // float4 index
    const int row  = i4 >> 6;                          // 64 float4 per row
    const int d    = (i4 & 63) * 4;
    const int code = codes[row];
    const float4 zv = ((const float4*)z)[i4];
    float4 q, o;
    q.x = emb[(size_t)(d + 0) * K_CODES + code];
    q.y = emb[(size_t)(d + 1) * K_CODES + code];
    q.z = emb[(size_t)(d + 2) * K_CODES + code];
    q.w = emb[(size_t)(d + 3) * K_CODES + code];
    o.x = q.x + (zv.x - q.x);
    o.y = q.y + (zv.y - q.y);
    o.z = q.z + (zv.z - q.z);
    o.w = q.w + (zv.w - q.w);
    ((float4*)zst)[i4] = o;
}

// ---------------------------------------------------------------------------
// Single-block scalar stats: loss (fixed-order sum of 512 partials),
// perplexity = exp(-sum p*log(p+1e-10)), usage = #nonzero codes.
// ---------------------------------------------------------------------------
__global__ __launch_bounds__(256) void vq_stats(
    const int* __restrict__ counts, const float* __restrict__ partial,
    float* __restrict__ out) {
    __shared__ float s_ent[256];
    __shared__ float s_use[256];
    const int tid = threadIdx.x;
    float ent = 0.f, use = 0.f;
    for (int k = tid; k < K_CODES; k += 256) {
        float p = (float)counts[k] * (1.0f / (float)N_ROWS);
        ent += p * logf(p + 1e-10f);
        if (p > 0.f) use += 1.f;
    }
    s_ent[tid] = ent;
    s_use[tid] = use;
    __syncthreads();
    for (int s = 128; s > 0; s >>= 1) {
        if (tid < s) { s_ent[tid] += s_ent[tid + s]; s_use[tid] += s_use[tid + s]; }
        __syncthreads();
    }
    if (tid == 0) {
        float tot = 0.f;
        for (int b = 0; b < NUM_BLOCKS; ++b) tot += partial[b];  // fixed order
        out[LOSS_OFF] = 0.25f * tot * (1.0f / ((float)N_ROWS * (float)D_DIM));
        out[PPL_OFF]  = expf(-s_ent[0]);
        out[USE_OFF]  = s_use[0];
    }
}

// ---------------------------------------------------------------------------
extern "C" void kernel_launch(void* const* d_in, const int* in_sizes, int n_in,
                              void* d_out, int out_size, void* d_ws, size_t ws_size,
                              hipStream_t stream) {
    const float* z   = (const float*)d_in[0];   // [16,4096,256] f32
    const float* emb = (const float*)d_in[1];   // [256,8192]    f32
    float* out = (float*)d_out;

    // workspace layout
    int*   ws_codes   = (int*)d_ws;                       // N_ROWS ints
    int*   ws_counts  = ws_codes + N_ROWS;                // K_CODES ints
    float* ws_esq     = (float*)(ws_counts + K_CODES);    // K_CODES floats
    float* ws_partial = ws_esq + K_CODES;                 // NUM_BLOCKS floats

    vq_zero_counts<<<K_CODES / 256, 256, 0, stream>>>(ws_counts);
    vq_esq<<<K_CODES / 256, 256, 0, stream>>>(emb, ws_esq);
    vq_argmin_wmma<<<NUM_BLOCKS, 256, 0, stream>>>(
        z, emb, ws_esq, ws_codes, ws_counts, ws_partial, out + IDX_OFF);
    vq_writeback<<<(N_ROWS * (D_DIM / 4)) / 256, 256, 0, stream>>>(
        z, emb, ws_codes, out + ZST_OFF);
    vq_stats<<<1, 256, 0, stream>>>(ws_counts, ws_partial, out);
}